// Icon_41850161332741
// MI455X (gfx1250) — compile-verified
//
#include <hip/hip_runtime.h>

// ---- problem constants (match reference) ----
#define HD 256          // H * D
#define NHEADS 4
#define DHEAD 64

typedef __attribute__((ext_vector_type(2))) float v2f;
typedef __attribute__((ext_vector_type(8))) float v8f;
typedef __attribute__((ext_vector_type(4))) unsigned int u32x4;
typedef __attribute__((ext_vector_type(4))) int i32x4;
typedef __attribute__((ext_vector_type(8))) int i32x8;

// ---------------------------------------------------------------------------
// out[n, c] = bias[c]   (d_out is poisoned by the harness; atomics add on top)
// ---------------------------------------------------------------------------
__global__ __launch_bounds__(256) void init_out_kernel(float* __restrict__ out,
                                                       const float* __restrict__ bias,
                                                       int total) {
  int i = blockIdx.x * blockDim.x + threadIdx.x;
  if (i < total) out[i] = bias[i & (HD - 1)];
}

// ---------------------------------------------------------------------------
// Build TDM descriptor group 0: {count=1, lds_addr, global_addr[56:0], type=2}
// (CDNA5 ISA 8.3)
// ---------------------------------------------------------------------------
__device__ __forceinline__ u32x4 tdm_group0(unsigned lds_addr, const void* gptr) {
  unsigned long long ga = (unsigned long long)gptr;
  u32x4 g0;
  g0.x = 0x1u;                                   // count=1, is_restore=0, no gather
  g0.y = lds_addr;                               // LDS byte address
  g0.z = (unsigned)ga;                           // global_addr[31:0]
  g0.w = ((unsigned)(ga >> 32) & 0x01FFFFFFu)    // global_addr[56:32]
       | 0x80000000u;                            // type=2 ("image") in bits[127:126]
  return g0;
}

// ---------------------------------------------------------------------------
// Build TDM descriptor group 1 for a 2-D f32 tile (CDNA5 ISA 8.4):
//   data_size=4B, pad_enable, pad_interval/pad_amount (LDS row padding),
//   tensor_dim0/1 (OOB clip -> zero fill), tile_dim0/1, tensor_dim0_stride.
// ---------------------------------------------------------------------------
__device__ __forceinline__ i32x8 tdm_group1_2d(unsigned pad_interval,  // 0=2,1=4,...,4=32,...,7=256 DW
                                               unsigned pad_amount,    // 0=1,...,3=4 DW
                                               unsigned tensor_dim0, unsigned tensor_dim1,
                                               unsigned tile_dim0, unsigned tile_dim1,
                                               unsigned long long dim0_stride) {
  i32x8 g1;
  g1[0] = (int)((2u << 16)                 // data_size = 4 bytes
              | (1u << 20)                 // pad_enable
              | (pad_interval << 22)
              | (pad_amount << 25));
  g1[1] = (int)((tensor_dim0 & 0xFFFFu) << 16);                 // [47:32]=abar addr(0), [63:48]=tdim0.lo
  g1[2] = (int)((tensor_dim0 >> 16) | ((tensor_dim1 & 0xFFFFu) << 16));
  g1[3] = (int)((tensor_dim1 >> 16) | (tile_dim0 << 16));
  g1[4] = (int)(tile_dim1 & 0xFFFFu);                           // tile_dim2 = 0
  g1[5] = (int)(unsigned)(dim0_stride & 0xFFFFFFFFull);
  g1[6] = (int)(unsigned)((dim0_stride >> 32) & 0xFFFFu);       // dim1_stride = 0 (2-D)
  g1[7] = 0;
  return g1;
}

// ---------------------------------------------------------------------------
// xp = x @ W  via V_WMMA_F32_16X16X4_F32, tiles staged by the Tensor Data
// Mover (TENSOR_LOAD_TO_LDS) with hardware LDS row padding.
//
// Block = 256 threads = 8 waves. Block tile: 64 rows x 256 cols.
// Per 32-wide k-chunk:
//   TDM #1: x[m0..m0+63][k0..k0+31] -> xs[64][36] (pad 4 DW / 32 DW)
//           tensor_dim1 = nRows-m0 -> hardware zero-fill of the M tail
//   TDM #2: W[k0..k0+31][0..255]    -> ws[32][260] (pad 4 DW / 256 DW)
//   wave 0 issues both + s_wait_tensorcnt 0; barrier publishes tiles.
//   each wave: 8 k-steps x (4 M-tiles x 2 N-tiles) = 64 WMMAs per chunk.
//
// VGPR layouts per CDNA5 ISA 7.12.2:
//   A 16x4 : lanes 0-15 -> M=lane, VGPR{0,1}=K{0,1}; lanes 16-31 = K{2,3}
//   B 4x16 : lanes 0-15 -> N=lane, VGPR{0,1}=K{0,1}; lanes 16-31 = K{2,3}
//   C/D    : VGPR r -> M=r (lanes 0-15) / M=r+8 (lanes 16-31), N=lane%16
// ---------------------------------------------------------------------------
__global__ __launch_bounds__(256) void gemm_wmma_f32(const float* __restrict__ x,
                                                     const float* __restrict__ w,
                                                     float* __restrict__ xp,
                                                     int nRows) {
  __shared__ float xs[64][36];    //  9,216 B (TDM-padded: 32 DW data + 4 DW pad)
  __shared__ float ws[32][260];   // 33,280 B (TDM-padded: 256 DW data + 4 DW pad)

  const int tid  = threadIdx.x;
  const int wave = tid >> 5;
  const int lane = tid & 31;
  const int half = lane >> 4;     // 0: K pair {0,1}, 1: K pair {2,3}
  const int l    = lane & 15;     // M index for A, N index for B/C/D

  const int m0 = blockIdx.x * 64;
  const int n0 = wave * 32;

  const unsigned lds_xs = (unsigned)(unsigned long long)&xs[0][0];
  const unsigned lds_ws = (unsigned)(unsigned long long)&ws[0][0];

  v8f c[4][2] = {};               // 8 accumulators = 64 VGPRs

  for (int k0 = 0; k0 < HD; k0 += 32) {
    __syncthreads();              // previous chunk fully consumed

    if (wave == 0) {              // uniform wave-level DMA issue (EXEC ignored)
      const i32x4 z4 = {0, 0, 0, 0};
      const i32x8 z8 = {0, 0, 0, 0, 0, 0, 0, 0};
      // x tile: 64 rows x 32 floats, row stride HD; zero-fill rows >= nRows
      {
        u32x4 g0 = tdm_group0(lds_xs, x + (size_t)m0 * HD + k0);
        i32x8 g1 = tdm_group1_2d(/*pad_int(32DW)=*/4, /*pad_amt(4DW)=*/3,
                                 /*tdim0=*/32, /*tdim1=*/(unsigned)(nRows - m0),
                                 /*tile0=*/32, /*tile1=*/64,
                                 /*stride0=*/(unsigned long long)HD);
        __builtin_amdgcn_tensor_load_to_lds(g0, g1, z4, z4, z8, 0);
      }
      // W tile: 32 rows x 256 floats, row stride HD
      {
        u32x4 g0 = tdm_group0(lds_ws, w + (size_t)k0 * HD);
        i32x8 g1 = tdm_group1_2d(/*pad_int(256DW)=*/7, /*pad_amt(4DW)=*/3,
                                 /*tdim0=*/HD, /*tdim1=*/32,
                                 /*tile0=*/HD, /*tile1=*/32,
                                 /*stride0=*/(unsigned long long)HD);
        __builtin_amdgcn_tensor_load_to_lds(g0, g1, z4, z4, z8, 0);
      }
      __builtin_amdgcn_s_wait_tensorcnt(0);
    }
    __syncthreads();              // publish TDM-written tiles to all waves

    // ---- compute: 8 k-steps x 8 WMMAs ----
#pragma unroll
    for (int ks = 0; ks < 8; ++ks) {
      const int kb = ks * 4 + 2 * half;
      v2f b0, b1;
      b0.x = ws[kb][n0 + l];        b0.y = ws[kb + 1][n0 + l];
      b1.x = ws[kb][n0 + 16 + l];   b1.y = ws[kb + 1][n0 + 16 + l];
#pragma unroll
      for (int mt = 0; mt < 4; ++mt) {
        v2f a = *(const v2f*)&xs[16 * mt + l][kb];   // 8B aligned (kb even)
        // 8 args: (neg_a, A, neg_b, B, c_mod, C, reuse_a, reuse_b)
        c[mt][0] = __builtin_amdgcn_wmma_f32_16x16x4_f32(false, a, false, b0,
                                                         (short)0, c[mt][0],
                                                         false, false);
        c[mt][1] = __builtin_amdgcn_wmma_f32_16x16x4_f32(false, a, false, b1,
                                                         (short)0, c[mt][1],
                                                         false, false);
      }
    }
  }

  // ---- store 64x32 result slab per wave (mask the M tail) ----
#pragma unroll
  for (int mt = 0; mt < 4; ++mt) {
#pragma unroll
    for (int r = 0; r < 8; ++r) {
      const int row = m0 + 16 * mt + r + 8 * half;
      if (row < nRows) {
        float* __restrict__ o = xp + (size_t)row * HD + n0 + l;
        o[0]  = c[mt][0][r];
        o[16] = c[mt][1][r];
      }
    }
  }
}

// ---------------------------------------------------------------------------
// Scatter phase: persistent grid, thread t in [0,256) owns feature t of each
// edge:  out[dst, t] += xp[src, t] * (net_w * attn[e, t/64]).
// xp (51 MB) + out (51 MB) fit in the 192 MB L2 -> gather + atomics L2-bound.
// 2x unrolled for memory-level parallelism; atomics are non-returning.
// ---------------------------------------------------------------------------
__global__ __launch_bounds__(256) void scatter_edges(const float* __restrict__ xp,
                                                     const int* __restrict__ edges,   // [2, E]
                                                     const float* __restrict__ attn,  // [E, H]
                                                     const float* __restrict__ net_w, // [3]
                                                     int net_idx,
                                                     float* __restrict__ out,
                                                     int E) {
  const int t = threadIdx.x;          // feature index 0..255
  const int h = t >> 6;               // head index 0..3
  const float wnet = net_w[net_idx];
  const int stride = gridDim.x;

  int e = blockIdx.x;
  for (; e + stride < E; e += 2 * stride) {
    const int e1 = e + stride;
    const int s0 = edges[e],      d0 = edges[E + e];
    const int s1 = edges[e1],     d1 = edges[E + e1];

    // warm L2 for the next iteration's first source row
    const int en = e + 2 * stride;
    if (en < E) __builtin_prefetch(xp + (size_t)edges[en] * HD + t, 0, 1);

    const float a0 = attn[(size_t)e  * NHEADS + h] * wnet;
    const float a1 = attn[(size_t)e1 * NHEADS + h] * wnet;
    const float v0 = xp[(size_t)s0 * HD + t] * a0;
    const float v1 = xp[(size_t)s1 * HD + t] * a1;
    atomicAdd(out + (size_t)d0 * HD + t, v0);
    atomicAdd(out + (size_t)d1 * HD + t, v1);
  }
  if (e < E) {
    const int s0 = edges[e], d0 = edges[E + e];
    const float a0 = attn[(size_t)e * NHEADS + h] * wnet;
    atomicAdd(out + (size_t)d0 * HD + t, xp[(size_t)s0 * HD + t] * a0);
  }
}

// ---------------------------------------------------------------------------
extern "C" void kernel_launch(void* const* d_in, const int* in_sizes, int n_in,
                              void* d_out, int out_size, void* d_ws, size_t ws_size,
                              hipStream_t stream) {
  const float* x    = (const float*)d_in[0];
  const int*   ei0  = (const int*)  d_in[1];
  const int*   ei1  = (const int*)  d_in[2];
  const int*   ei2  = (const int*)  d_in[3];
  const float* at0  = (const float*)d_in[4];
  const float* at1  = (const float*)d_in[5];
  const float* at2  = (const float*)d_in[6];
  const float* netw = (const float*)d_in[7];
  const float* lw   = (const float*)d_in[8];
  const float* bias = (const float*)d_in[9];
  float* out = (float*)d_out;

  const int nRows = in_sizes[0] / HD;   // N
  const int E     = in_sizes[1] / 2;    // edges per network

  // Workspace: xp = projected features, nRows * HD floats (~51.2 MB).
  float* xp = (float*)d_ws;

  // 1) out = bias (broadcast)
  {
    const int total = nRows * HD;
    init_out_kernel<<<(total + 255) / 256, 256, 0, stream>>>(out, bias, total);
  }

  // 2) xp = x @ W  (fp32 WMMA, TDM-staged LDS tiles)
  {
    const int mBlocks = (nRows + 63) / 64;
    gemm_wmma_f32<<<mBlocks, 256, 0, stream>>>(x, lw, xp, nRows);
  }

  // 3) edge scatter, one launch per network (stream-ordered after GEMM)
  const int SCATTER_BLOCKS = 4096;
  scatter_edges<<<SCATTER_BLOCKS, 256, 0, stream>>>(xp, ei0, at0, netw, 0, out, E);
  scatter_edges<<<SCATTER_BLOCKS, 256, 0, stream>>>(xp, ei1, at1, netw, 1, out, E);
  scatter_edges<<<SCATTER_BLOCKS, 256, 0, stream>>>(xp, ei2, at2, netw, 2, out, E);
}